// GCN_6657199308988
// MI455X (gfx1250) — compile-verified
//
#include <hip/hip_runtime.h>

#define N_NODES 50000
#define N_EDGES 500000
#define IN_F 128
#define HID 256
#define EMB 128

typedef float v2f __attribute__((ext_vector_type(2)));
typedef float v8f __attribute__((ext_vector_type(8)));

// ---------------------------------------------------------------------------
// Zero-fill (float4 granularity; all our buffers are multiples of 4 floats)
// ---------------------------------------------------------------------------
__global__ void gcn_zero_kernel(float4* __restrict__ p, long n4) {
    long i = blockIdx.x * (long)blockDim.x + threadIdx.x;
    const long stride = (long)gridDim.x * blockDim.x;
    float4 z; z.x = z.y = z.z = z.w = 0.f;
    for (; i < n4; i += stride) p[i] = z;
}

// ---------------------------------------------------------------------------
// Edge scatter-add: agg[dst[e], :] += h[src[e], :]
// One thread handles one float4 chunk of one edge's feature row.
// Consecutive threads cover consecutive chunks -> coalesced 512B/wave reads;
// atomics resolve in L2 (feature matrices fit in the 192MB L2).
// ---------------------------------------------------------------------------
template <int F>
__global__ void gcn_scatter_add_kernel(const float* __restrict__ h,
                                       const int* __restrict__ src,
                                       const int* __restrict__ dst,
                                       float* __restrict__ agg,
                                       int nEdges) {
    constexpr int F4 = F / 4;   // power of two -> shifts, no division
    long tid = blockIdx.x * (long)blockDim.x + threadIdx.x;
    int e = (int)(tid / F4);
    int j = (int)(tid % F4);
    if (e >= nEdges) return;
    const int s = src[e];
    const int d = dst[e];
    const float4 v = *(const float4*)(h + (long)s * F + j * 4);
    float* p = agg + (long)d * F + j * 4;
    atomicAdd(p + 0, v.x);
    atomicAdd(p + 1, v.y);
    atomicAdd(p + 2, v.z);
    atomicAdd(p + 3, v.w);
}

// ---------------------------------------------------------------------------
// FP32 WMMA GEMM: out = op(A[N,K] @ W[K,Fout] (+ bias) (relu))
// One wave computes a 16x16 tile via V_WMMA_F32_16X16X4_F32.
// Block = 256 threads = 8 waves sharing the same 16 A-rows (L0 reuse),
// covering 8 consecutive column tiles.
// Requires: N % 16 == 0, Fout % 128 == 0 (true for N=50000, Fout in {128,256})
// so there is no divergence and EXEC is all-1s around the WMMA.
//
// Operand layouts per CDNA5 ISA 7.12.2:
//   A 16x4 f32 : lane L (L<16): M=L, holds K=0,1 in v0,v1; lanes 16-31: K=2,3
//   B 4x16 f32 : lane L (L<16): N=L, holds K=0,1;          lanes 16-31: K=2,3
//   C/D 16x16  : VGPR r, lanes 0-15 -> M=r, lanes 16-31 -> M=r+8; col = lane%16
// ---------------------------------------------------------------------------
template <bool RELU, bool BIAS>
__global__ void gcn_gemm_wmma_kernel(const float* __restrict__ A,
                                     const float* __restrict__ W,
                                     const float* __restrict__ bias,
                                     float* __restrict__ out,
                                     int N, int K, int Fout) {
    const int lane = threadIdx.x & 31;
    const int wave = threadIdx.x >> 5;
    const int lid  = lane & 15;
    const int half = lane >> 4;

    const int row0 = blockIdx.x * 16;
    const int col0 = (blockIdx.y * 8 + wave) * 16;

    const float* __restrict__ arow = A + (long)(row0 + lid) * K;
    const float* __restrict__ wcol = W + col0 + lid;

    v8f c = {};
    for (int kk = 0; kk < K; kk += 4) {
        const int ka = kk + 2 * half;              // even -> 8B aligned
        v2f a = *(const v2f*)(arow + ka);          // A[M=lid][ka], A[M=lid][ka+1]
        v2f b;
        b.x = wcol[(long)ka * Fout];               // W[ka  ][col0+lid]
        b.y = wcol[(long)(ka + 1) * Fout];         // W[ka+1][col0+lid]
        c = __builtin_amdgcn_wmma_f32_16x16x4_f32(
                /*neg_a=*/false, a, /*neg_b=*/false, b,
                /*c_mod=*/(short)0, c, /*reuse_a=*/false, /*reuse_b=*/false);
    }

    const float bv = BIAS ? bias[col0 + lid] : 0.f;
#pragma unroll
    for (int r = 0; r < 8; ++r) {
        float v = c[r] + bv;
        if (RELU) v = fmaxf(v, 0.f);
        out[(long)(row0 + r + 8 * half) * Fout + col0 + lid] = v;
    }
}

// ---------------------------------------------------------------------------
// Final: out[f] = mean_n relu(agg3[n,f] + b3[f]),  f in [0,128)
// Block of 128 threads (thread = feature), grid-stride over nodes,
// per-thread partial sum then one float atomic per (block,feature).
// ---------------------------------------------------------------------------
__global__ void gcn_final_mean_kernel(const float* __restrict__ agg,
                                      const float* __restrict__ b3,
                                      float* __restrict__ out,
                                      int N) {
    const int f = threadIdx.x;           // 128 threads
    const float bv = b3[f];
    float sum = 0.f;
    for (int n = blockIdx.x; n < N; n += gridDim.x)
        sum += fmaxf(agg[(long)n * EMB + f] + bv, 0.f);
    atomicAdd(&out[f], sum * (1.0f / (float)N));
}

// ---------------------------------------------------------------------------
// Orchestration. Uses the linearity of segment_sum:
//   agg(h) @ W == agg(h @ W)
// Layer 1: aggregate 128 feats, then GEMM to 256.
// Layer 2: aggregate 256, GEMM 256.
// Layer 3: GEMM 256->128 FIRST (no bias), then aggregate only 128 feats,
//          add bias + relu + mean in the reduction.
// Workspace: A = ws[0 .. N*256), B = ws[N*256 .. 2*N*256)  (102.4 MB)
// ---------------------------------------------------------------------------
extern "C" void kernel_launch(void* const* d_in, const int* in_sizes, int n_in,
                              void* d_out, int out_size, void* d_ws, size_t ws_size,
                              hipStream_t stream) {
    const float* x  = (const float*)d_in[0];
    const int*   src= (const int*)  d_in[1];
    const int*   dst= (const int*)  d_in[2];
    const float* W1 = (const float*)d_in[3];
    const float* b1 = (const float*)d_in[4];
    const float* W2 = (const float*)d_in[5];
    const float* b2 = (const float*)d_in[6];
    const float* W3 = (const float*)d_in[7];
    const float* b3 = (const float*)d_in[8];
    float* out = (float*)d_out;

    float* A = (float*)d_ws;                       // capacity N*256 floats
    float* B = A + (long)N_NODES * HID;            // capacity N*256 floats

    const dim3 zgrid(2048), zblk(256);

    // ---- Layer 1: agg1 = scatter(x) [N,128]; B = relu(agg1@W1 + b1) [N,256]
    gcn_zero_kernel<<<zgrid, zblk, 0, stream>>>((float4*)A, (long)N_NODES * IN_F / 4);
    {
        long total = (long)N_EDGES * (IN_F / 4);
        gcn_scatter_add_kernel<IN_F><<<(int)((total + 255) / 256), 256, 0, stream>>>(
            x, src, dst, A, N_EDGES);
    }
    gcn_gemm_wmma_kernel<true, true><<<dim3(N_NODES / 16, HID / 128), 256, 0, stream>>>(
        A, W1, b1, B, N_NODES, IN_F, HID);

    // ---- Layer 2: agg2 = scatter(B) [N,256]; B = relu(agg2@W2 + b2) [N,256]
    gcn_zero_kernel<<<zgrid, zblk, 0, stream>>>((float4*)A, (long)N_NODES * HID / 4);
    {
        long total = (long)N_EDGES * (HID / 4);
        gcn_scatter_add_kernel<HID><<<(int)((total + 255) / 256), 256, 0, stream>>>(
            B, src, dst, A, N_EDGES);
    }
    gcn_gemm_wmma_kernel<true, true><<<dim3(N_NODES / 16, HID / 128), 256, 0, stream>>>(
        A, W2, b2, B, N_NODES, HID, HID);

    // ---- Layer 3 (transformed): A = B@W3 [N,128] (no bias/relu yet),
    //      then agg3 = scatter(A) into B [N,128]
    gcn_gemm_wmma_kernel<false, false><<<dim3(N_NODES / 16, EMB / 128), 256, 0, stream>>>(
        B, W3, nullptr, A, N_NODES, HID, EMB);
    gcn_zero_kernel<<<zgrid, zblk, 0, stream>>>((float4*)B, (long)N_NODES * EMB / 4);
    {
        long total = (long)N_EDGES * (EMB / 4);
        gcn_scatter_add_kernel<EMB><<<(int)((total + 255) / 256), 256, 0, stream>>>(
            A, src, dst, B, N_EDGES);
    }

    // ---- out = mean_n relu(agg3 + b3)
    gcn_zero_kernel<<<1, 32, 0, stream>>>((float4*)out, EMB / 4);
    gcn_final_mean_kernel<<<512, EMB, 0, stream>>>(B, b3, out, N_NODES);
}